// GCNTree_83451214561512
// MI455X (gfx1250) — compile-verified
//
#include <hip/hip_runtime.h>
#include <math.h>

// ---------------------------------------------------------------------------
// GCN + soft-decision-tree head for MI455X (gfx1250, wave32, WMMA).
// Dominant cost: two passes of dense A [8192x8192] fp32 (256MB each) times a
// 16-wide panel -> memory bound (~6 FLOP/B). Implemented as fp32 WMMA
// 16x16x4 rank updates streaming A at full bandwidth.
// ---------------------------------------------------------------------------

typedef __attribute__((ext_vector_type(2))) float v2f;
typedef __attribute__((ext_vector_type(8))) float v8f;

#define NN 8192
#define FF 256
#define KS 4                 // split-K factor for the big GEMM
#define SEG (NN / KS)        // 2048 K per wave
#define STRIPS (NN / 16)     // 512 row strips

// ---------------- k1: Bt1[16][N] = (X @ W1)^T --------------------------------
__global__ void __launch_bounds__(256)
xw1t_kernel(const float* __restrict__ X, const float* __restrict__ W1,
            float* __restrict__ Bt) {
  int tid = blockIdx.x * blockDim.x + threadIdx.x;   // 131072 threads
  int row = tid >> 4;
  int col = tid & 15;
  const float* xr = X + (size_t)row * FF;
  float acc = 0.f;
#pragma unroll 8
  for (int j = 0; j < FF; ++j)
    acc = fmaf(xr[j], W1[j * 16 + col], acc);
  Bt[(size_t)col * NN + row] = acc;
}

// ---------------- big GEMM: P[g][strip] += A[strip rows] @ B (16 cols) -------
// A-frag layout (V_WMMA_F32_16X16X4_F32): lanes 0-15 = rows M, VGPR0/1 = K0/K1;
// lanes 16-31 = rows M, VGPR0/1 = K2/K3.  We permute the K-slot->column map
// (sum over K commutes) so a single float4 per lane feeds two WMMAs, and the
// matching B fragments come from one contiguous float4 of the transposed panel.
__global__ void __launch_bounds__(256)
gemm_strip_kernel(const float* __restrict__ A, const float* __restrict__ Bt,
                  float* __restrict__ P) {
  const int lane  = threadIdx.x & 31;
  const int wave  = threadIdx.x >> 5;
  const int strip = blockIdx.x * 8 + wave;   // 0..511
  const int g     = blockIdx.y;              // 0..KS-1
  const int hi    = lane >> 4;               // 0 or 1
  const int m     = lane & 15;               // A row within strip == B column

  const float* __restrict__ arow = A  + (size_t)(strip * 16 + m) * NN + g * SEG + 4 * hi;
  const float* __restrict__ brow = Bt + (size_t)m * NN              + g * SEG + 4 * hi;

  v8f c = {};
  for (int k0 = 0; k0 < SEG; k0 += 128) {
    // prefetch the A stream ~8KB ahead (guarded: stay inside A allocation)
    if (g * SEG + k0 + 2048 + 8 <= NN) {
      __builtin_prefetch(arow + k0 + 2048, 0, 0);
    }
#pragma unroll
    for (int k = 0; k < 128; k += 8) {
      const float4 a4 = *(const float4*)(arow + k0 + k);
      const float4 b4 = *(const float4*)(brow + k0 + k);
      v2f a0 = {a4.x, a4.y};
      v2f b0 = {b4.x, b4.y};
      v2f a1 = {a4.z, a4.w};
      v2f b1 = {b4.z, b4.w};
      c = __builtin_amdgcn_wmma_f32_16x16x4_f32(false, a0, false, b0,
                                                (short)0, c, false, false);
      c = __builtin_amdgcn_wmma_f32_16x16x4_f32(false, a1, false, b1,
                                                (short)0, c, false, false);
    }
  }

  // C/D layout: VGPR i -> row i (lanes 0-15) / row i+8 (lanes 16-31), col = m
  float* __restrict__ p = P + ((size_t)g * STRIPS + strip) * 256;
#pragma unroll
  for (int i = 0; i < 8; ++i)
    p[(hi ? i + 8 : i) * 16 + m] = c[i];
}

// ---------------- epilogue: Out[n][c] = relu(sum_g P + bias) -----------------
__global__ void __launch_bounds__(256)
epilogue_kernel(const float* __restrict__ P, const float* __restrict__ bias,
                int bias_n, float* __restrict__ Out) {
  int tid = blockIdx.x * blockDim.x + threadIdx.x;   // 131072 threads
  int n = tid >> 4;
  int cidx = tid & 15;
  int strip = n >> 4;
  int i = n & 15;
  size_t e = (size_t)strip * 256 + i * 16 + cidx;
  float s = 0.f;
#pragma unroll
  for (int gg = 0; gg < KS; ++gg)
    s += P[(size_t)gg * STRIPS * 256 + e];
  s += (cidx < bias_n) ? bias[cidx] : 0.f;
  Out[(size_t)n * 16 + cidx] = s > 0.f ? s : 0.f;
}

// ---------------- k3: Bt2[16][N] = (H @ W2)^T, zero-padded to 16 cols --------
__global__ void __launch_bounds__(256)
hw2t_kernel(const float* __restrict__ H, const float* __restrict__ W2,
            float* __restrict__ Bt2) {
  int tid = blockIdx.x * blockDim.x + threadIdx.x;   // 131072 threads
  int n = tid >> 4;
  int cidx = tid & 15;
  float acc = 0.f;
  if (cidx < 8) {
    const float* hr = H + (size_t)n * 16;
#pragma unroll
    for (int j = 0; j < 16; ++j)
      acc = fmaf(hr[j], W2[j * 8 + cidx], acc);
  }
  Bt2[(size_t)cidx * NN + n] = acc;
}

// ---------------- tail: decision tree head, one thread per node --------------
__global__ void __launch_bounds__(256)
tail_kernel(const float* __restrict__ Z, const float* __restrict__ Wd,
            const float* __restrict__ bd, const float* __restrict__ mask,
            const float* __restrict__ Wdec, const float* __restrict__ bdec,
            const float* __restrict__ pi, float* __restrict__ Out) {
  int n = blockIdx.x * blockDim.x + threadIdx.x;
  if (n >= NN) return;
  const float* zr = Z + (size_t)n * 16;   // only cols 0..7 are live (L=8)

  float zv[8];
#pragma unroll
  for (int j = 0; j < 8; ++j) zv[j] = zr[j];

  // feat = relu(z @ Wd + bd)
  float feat[16];
#pragma unroll
  for (int l = 0; l < 16; ++l) {
    float a = bd[l];
#pragma unroll
    for (int j = 0; j < 8; ++j) a = fmaf(zv[j], Wd[j * 16 + l], a);
    feat[l] = a > 0.f ? a : 0.f;
  }

  // feature permutation: fp = feat @ mask
  float fp[16];
#pragma unroll
  for (int l = 0; l < 16; ++l) {
    float a = 0.f;
#pragma unroll
    for (int i = 0; i < 16; ++i) a = fmaf(feat[i], mask[i * 16 + l], a);
    fp[l] = a;
  }

  // Only the last tree level contributes (faithful to reference):
  // mu[2j]   = fp[8+j] * sigmoid((fp@Wdec+bdec)[8+j])
  // mu[2j+1] = fp[8+j] * (1 - sigmoid(...))
  float out[10];
#pragma unroll
  for (int cc = 0; cc < 10; ++cc) out[cc] = 0.f;

#pragma unroll
  for (int j = 0; j < 8; ++j) {
    const int col = 8 + j;
    float x = bdec[col];
#pragma unroll
    for (int i = 0; i < 16; ++i) x = fmaf(fp[i], Wdec[i * 16 + col], x);
    const float s  = 1.f / (1.f + expf(-x));
    const float f  = fp[col];
    const float m0 = f * s;
    const float m1 = f - m0;          // f * (1 - s)
#pragma unroll
    for (int cc = 0; cc < 10; ++cc) {
      float p0 = pi[(2 * j) * 10 + cc];     p0 = p0 > 0.f ? p0 : 0.f;
      float p1 = pi[(2 * j + 1) * 10 + cc]; p1 = p1 > 0.f ? p1 : 0.f;
      out[cc] = fmaf(m0, p0, fmaf(m1, p1, out[cc]));
    }
  }
#pragma unroll
  for (int cc = 0; cc < 10; ++cc)
    Out[(size_t)n * 10 + cc] = out[cc];
}

// ---------------------------------------------------------------------------
extern "C" void kernel_launch(void* const* d_in, const int* in_sizes, int n_in,
                              void* d_out, int out_size, void* d_ws, size_t ws_size,
                              hipStream_t stream) {
  (void)in_sizes; (void)n_in; (void)out_size; (void)ws_size;
  const float* X    = (const float*)d_in[0];
  const float* A    = (const float*)d_in[1];
  const float* W1   = (const float*)d_in[2];
  const float* b1   = (const float*)d_in[3];
  const float* W2   = (const float*)d_in[4];
  const float* b2   = (const float*)d_in[5];
  const float* Wd   = (const float*)d_in[6];
  const float* bd   = (const float*)d_in[7];
  const float* mask = (const float*)d_in[8];
  const float* Wdec = (const float*)d_in[9];
  const float* bdec = (const float*)d_in[10];
  const float* pi   = (const float*)d_in[11];
  float* out = (float*)d_out;

  // ws layout (floats): 4 MB total
  float* ws  = (float*)d_ws;
  float* Bt1 = ws;                 // 131072  (X@W1)^T
  float* P   = ws + 131072;        // 524288  split-K partials
  float* H   = ws + 655360;        // 131072  relu(A@XW1+b1), row-major [N,16]
  float* Bt2 = ws + 786432;        // 131072  (H@W2)^T padded to 16
  float* Z   = ws + 917504;        // 131072  relu(A@HW2+b2), row-major [N,16]

  xw1t_kernel<<<512, 256, 0, stream>>>(X, W1, Bt1);
  gemm_strip_kernel<<<dim3(STRIPS / 8, KS), 256, 0, stream>>>(A, Bt1, P);
  epilogue_kernel<<<512, 256, 0, stream>>>(P, b1, 16, H);
  hw2t_kernel<<<512, 256, 0, stream>>>(H, W2, Bt2);
  gemm_strip_kernel<<<dim3(STRIPS / 8, KS), 256, 0, stream>>>(A, Bt2, P);
  epilogue_kernel<<<512, 256, 0, stream>>>(P, b2, 8, Z);
  tail_kernel<<<NN / 256, 256, 0, stream>>>(Z, Wd, bd, mask, Wdec, bdec, pi, out);
}